// SAN_79826262163504
// MI455X (gfx1250) — compile-verified
//
#include <hip/hip_runtime.h>
#include <hip/hip_bf16.h>

typedef float v2f __attribute__((ext_vector_type(2)));
typedef float v8f __attribute__((ext_vector_type(8)));

#define TM 128
#define TN 128
#define TK 32
#define KP (TK + 4)          // padded K stride: 36 floats -> conflict-free frags
#define ALPHA_LRELU 0.2f
#define EPS_P 0.1f
#define NEG_MASK -9.0e15f

// ---------------------------------------------------------------------------
// Generic fp32 WMMA GEMM: C[M,N] = alpha * op(A)[M,K] * op(B)[K,N] + (S ? S : 0)
// op selected by transA/transB; lda/ldb are PHYSICAL row strides.
//   transA==0: A(m,k) = A[m*lda + k]      transA==1: A(m,k) = A[k*lda + m]
//   transB==0: B(k,n) = B[k*ldb + n]      transB==1: B(k,n) = B[n*ldb + k]
// Requires M%128==0, N%128==0, K%32==0. Block = 256 threads = 8 waves (wave32)
// arranged 4x2; each wave owns a 32x64 tile = 2x4 V_WMMA_F32_16X16X4_F32 accs.
// Both A and B tiles are staged K-innermost in LDS so every WMMA fragment is
// one contiguous 8-byte LDS read (ds_load_b64), no cross-VGPR packing movs.
// ---------------------------------------------------------------------------
__global__ __launch_bounds__(256) void san_gemm_f32_wmma(
    float* __restrict__ C, const float* __restrict__ A, const float* __restrict__ B,
    const float* __restrict__ S, int M, int N, int K, int lda, int ldb,
    int transA, int transB, float alpha)
{
    __shared__ float As[TM][KP];   // As[m][k]
    __shared__ float Bs[TN][KP];   // Bs[n][k]  (K-innermost!)

    const int tid  = threadIdx.x;
    const int lane = tid & 31;
    const int wave = tid >> 5;
    const int wm   = wave >> 1;   // 0..3 (row group of 32)
    const int wn   = wave & 1;    // 0..1 (col group of 64)
    const int m0   = blockIdx.y * TM;
    const int n0   = blockIdx.x * TN;
    const int half = lane >> 4;   // selects K pair per ISA A/B fragment layout
    const int l15  = lane & 15;

    v8f acc[2][4];
#pragma unroll
    for (int mi = 0; mi < 2; ++mi)
#pragma unroll
        for (int ni = 0; ni < 4; ++ni)
#pragma unroll
            for (int e = 0; e < 8; ++e) acc[mi][ni][e] = 0.0f;

    for (int k0 = 0; k0 < K; k0 += TK) {
        // ---- stage A tile [TM x TK] into As[m][k] ----
        if (!transA) {
            for (int i = tid; i < (TM * TK / 4); i += 256) {   // 1024 float4
                int row = i >> 3;
                int c4  = (i & 7) << 2;
                const float4 v = *reinterpret_cast<const float4*>(
                    &A[(size_t)(m0 + row) * lda + k0 + c4]);
                As[row][c4 + 0] = v.x; As[row][c4 + 1] = v.y;
                As[row][c4 + 2] = v.z; As[row][c4 + 3] = v.w;
            }
        } else {
            for (int i = tid; i < TM * TK; i += 256) {
                int k = i >> 7;      // 0..31
                int m = i & 127;     // coalesced over physical row
                As[m][k] = A[(size_t)(k0 + k) * lda + m0 + m];
            }
        }
        // ---- stage B tile [TK x TN] into Bs[n][k] ----
        if (!transB) {
            for (int i = tid; i < (TK * TN / 4); i += 256) {   // 1024 float4
                int k  = i >> 5;            // 0..31
                int c4 = (i & 31) << 2;     // 0..124
                const float4 v = *reinterpret_cast<const float4*>(
                    &B[(size_t)(k0 + k) * ldb + n0 + c4]);
                Bs[c4 + 0][k] = v.x; Bs[c4 + 1][k] = v.y;
                Bs[c4 + 2][k] = v.z; Bs[c4 + 3][k] = v.w;
            }
        } else {
            for (int i = tid; i < TK * TN; i += 256) {
                int n = i >> 5;      // 0..127
                int k = i & 31;      // coalesced over physical row
                Bs[n][k] = B[(size_t)(n0 + n) * ldb + k0 + k];
            }
        }
        __syncthreads();

#pragma unroll
        for (int kk = 0; kk < TK; kk += 4) {
            const int kb = kk + (half << 1);
            v2f af[2], bf[4];
#pragma unroll
            for (int mi = 0; mi < 2; ++mi) {
                const int row = wm * 32 + mi * 16 + l15;
                af[mi].x = As[row][kb];
                af[mi].y = As[row][kb + 1];
            }
#pragma unroll
            for (int ni = 0; ni < 4; ++ni) {
                const int col = wn * 64 + ni * 16 + l15;
                bf[ni].x = Bs[col][kb];
                bf[ni].y = Bs[col][kb + 1];
            }
#pragma unroll
            for (int mi = 0; mi < 2; ++mi)
#pragma unroll
                for (int ni = 0; ni < 4; ++ni)
                    acc[mi][ni] = __builtin_amdgcn_wmma_f32_16x16x4_f32(
                        false, af[mi], false, bf[ni], (short)0, acc[mi][ni],
                        false, false);
        }
        __syncthreads();
    }

    // ---- epilogue: C layout = lanes0-15 -> M=v, lanes16-31 -> M=8+v; N=lane&15
#pragma unroll
    for (int mi = 0; mi < 2; ++mi) {
#pragma unroll
        for (int ni = 0; ni < 4; ++ni) {
            const int colb = n0 + wn * 64 + ni * 16 + l15;
            const int rowb = m0 + wm * 32 + mi * 16 + (half << 3);
#pragma unroll
            for (int v = 0; v < 8; ++v) {
                const size_t idx = (size_t)(rowb + v) * N + colb;
                float val = alpha * acc[mi][ni][v];
                if (S) val += S[idx];
                C[idx] = val;
            }
        }
    }
}

// ---------------------------------------------------------------------------
__global__ void san_zero(float* __restrict__ p, int n)
{
    int i = blockIdx.x * 256 + threadIdx.x;
    if (i < n) p[i] = 0.0f;
}

__global__ void san_scatter_add(float* __restrict__ Bm, const int* __restrict__ rows,
                                const int* __restrict__ cols, const float* __restrict__ vals,
                                int nnz, int ld)
{
    int i = blockIdx.x * 256 + threadIdx.x;
    if (i < nnz) atomicAdd(&Bm[(size_t)rows[i] * ld + cols[i]], vals[i]);
}

// a[i] = Xf[i,:] . att[0:128]; b[i] = Xf[i,:] . att[128:256]  (one wave per row)
__global__ void san_rowdot2(const float* __restrict__ Xf, const float* __restrict__ att,
                            float* __restrict__ a, float* __restrict__ b, int N)
{
    const int wave = threadIdx.x >> 5, lane = threadIdx.x & 31;
    const int row = blockIdx.x * 8 + wave;
    if (row >= N) return;
    const float* x = Xf + (size_t)row * 128;
    float sa = 0.0f, sb = 0.0f;
    for (int c = lane; c < 128; c += 32) {
        const float xv = x[c];
        sa += xv * att[c];
        sb += xv * att[128 + c];
    }
#pragma unroll
    for (int off = 16; off; off >>= 1) {
        sa += __shfl_xor(sa, off, 32);
        sb += __shfl_xor(sb, off, 32);
    }
    if (lane == 0) { a[row] = sa; b[row] = sb; }
}

// Row-wise masked leaky-relu softmax: Wt[i,j] = softmax_j( mask(L[i,j]!=0,
// leaky(a_i + b_j), NEG) ). All-masked rows naturally yield uniform 1/N.
__global__ __launch_bounds__(256) void san_att_softmax(
    float* __restrict__ Wt, const float* __restrict__ L,
    const float* __restrict__ a, const float* __restrict__ b, int N)
{
    __shared__ float red[256];
    const int i = blockIdx.x;
    const float ai = a[i];
    const float* Lrow = L + (size_t)i * N;
    float* Wrow = Wt + (size_t)i * N;

    float m = -3.4e38f;
    for (int j = threadIdx.x; j < N; j += 256) {
        const float t = ai + b[j];
        const float v = (t > 0.0f) ? t : ALPHA_LRELU * t;
        const float e = (Lrow[j] != 0.0f) ? v : NEG_MASK;
        m = fmaxf(m, e);
    }
    red[threadIdx.x] = m; __syncthreads();
    for (int s = 128; s; s >>= 1) {
        if (threadIdx.x < s) red[threadIdx.x] = fmaxf(red[threadIdx.x], red[threadIdx.x + s]);
        __syncthreads();
    }
    m = red[0]; __syncthreads();

    float sum = 0.0f;
    for (int j = threadIdx.x; j < N; j += 256) {
        const float t = ai + b[j];
        const float v = (t > 0.0f) ? t : ALPHA_LRELU * t;
        const float e = (Lrow[j] != 0.0f) ? v : NEG_MASK;
        const float w = __expf(e - m);
        Wrow[j] = w;
        sum += w;
    }
    red[threadIdx.x] = sum; __syncthreads();
    for (int s = 128; s; s >>= 1) {
        if (threadIdx.x < s) red[threadIdx.x] += red[threadIdx.x + s];
        __syncthreads();
    }
    const float inv = 1.0f / red[0];
    for (int j = threadIdx.x; j < N; j += 256) Wrow[j] *= inv;
}

__global__ void san_sigmoid(float* __restrict__ out, const float* __restrict__ in, int n)
{
    int i = blockIdx.x * 256 + threadIdx.x;
    if (i < n) out[i] = 1.0f / (1.0f + __expf(-in[i]));
}

// ---------------------------------------------------------------------------
extern "C" void kernel_launch(void* const* d_in, const int* in_sizes, int n_in,
                              void* d_out, int out_size, void* d_ws, size_t ws_size,
                              hipStream_t stream)
{
    (void)in_sizes; (void)n_in; (void)out_size; (void)ws_size;
    const int N0 = 1024, N1 = 4096, N2 = 1024, F = 128;

    const float* z0  = (const float*)d_in[0];
    const float* z1  = (const float*)d_in[1];
    const float* z2  = (const float*)d_in[2];
    const int*   b1i = (const int*)  d_in[3];   // [2, 2*N1] flat: rows then cols
    const float* b1v = (const float*)d_in[4];
    const int*   b2i = (const int*)  d_in[5];   // [2, 3*N2]
    const float* b2v = (const float*)d_in[6];
    const float* W   = (const float*)d_in[7];   // [128,128]
    const float* Am  = (const float*)d_in[8];
    const float* R   = (const float*)d_in[9];
    const float* Dm  = (const float*)d_in[10];
    const float* Wh  = (const float*)d_in[11];
    const float* at11 = (const float*)d_in[12]; // [256]
    const float* at12 = (const float*)d_in[13];
    const float* at13 = (const float*)d_in[14];
    const float* at21 = (const float*)d_in[15];
    const float* at22 = (const float*)d_in[16];
    float* out = (float*)d_out;

    // workspace layout (floats)
    float* p = (float*)d_ws;
    float* B1  = p; p += (size_t)N0 * N1;
    float* B2  = p; p += (size_t)N1 * N2;
    float* L1d = p; p += (size_t)N1 * N1;
    float* L1u = p; p += (size_t)N1 * N1;
    float* L0  = p; p += (size_t)N0 * N0;
    float* L2  = p; p += (size_t)N2 * N2;
    float* Wb  = p; p += (size_t)N1 * N1;   // attention weights (reused)
    float* x1d = p; p += (size_t)N1 * F;
    float* x2u = p; p += (size_t)N2 * F;
    float* Xf  = p; p += (size_t)N1 * F;
    float* av  = p; p += N1;
    float* bv  = p; p += N1;
    float* u   = p; p += (size_t)N1 * F;
    float* t1  = p; p += (size_t)N1 * F;
    float* t2  = p; p += (size_t)N1 * F;
    float* y0  = p; p += (size_t)N0 * F;
    float* y1  = p; p += (size_t)N1 * F;
    float* y2  = p; p += (size_t)N2 * F;

    auto gemm = [&](float* C, const float* A, const float* B, const float* S,
                    int M, int N, int K, int lda, int ldb, int tA, int tB, float alpha) {
        dim3 grid(N / TN, M / TM);
        san_gemm_f32_wmma<<<grid, 256, 0, stream>>>(C, A, B, S, M, N, K, lda, ldb, tA, tB, alpha);
    };

    // ---- build dense incidence matrices (re-zeroed every call: deterministic)
    san_zero<<<(N0 * N1 + 255) / 256, 256, 0, stream>>>(B1, N0 * N1);
    san_zero<<<(N1 * N2 + 255) / 256, 256, 0, stream>>>(B2, N1 * N2);
    san_scatter_add<<<(2 * N1 + 255) / 256, 256, 0, stream>>>(B1, b1i, b1i + 2 * N1, b1v, 2 * N1, N1);
    san_scatter_add<<<(3 * N2 + 255) / 256, 256, 0, stream>>>(B2, b2i, b2i + 3 * N2, b2v, 3 * N2, N2);

    // ---- Laplacians (exact fp32 via f32 WMMA)
    gemm(L1d, B1, B1, nullptr, N1, N1, N0, N1, N1, 1, 0, 1.0f);  // B1^T B1
    gemm(L1u, B2, B2, nullptr, N1, N1, N2, N2, N2, 0, 1, 1.0f);  // B2 B2^T
    gemm(L0,  B1, B1, nullptr, N0, N0, N1, N1, N1, 0, 1, 1.0f);  // B1 B1^T
    gemm(L2,  B2, B2, nullptr, N2, N2, N1, N2, N2, 1, 0, 1.0f);  // B2^T B2

    // ---- lifted features
    gemm(x1d, B1, z0, nullptr, N1, F, N0, N1, F, 1, 0, 1.0f);    // B1^T z0
    gemm(x2u, B2, z1, nullptr, N2, F, N1, N2, F, 1, 0, 1.0f);    // B2^T z1

    // ================= z1 pipeline =================
    // A11 = ef(z1, W, L1d, at11); y1 = (A11 z1) W
    gemm(Xf, z1, W, nullptr, N1, F, F, F, F, 0, 0, 1.0f);
    san_rowdot2<<<N1 / 8, 256, 0, stream>>>(Xf, at11, av, bv, N1);
    san_att_softmax<<<N1, 256, 0, stream>>>(Wb, L1d, av, bv, N1);
    gemm(u, Wb, z1, nullptr, N1, F, N1, N1, F, 0, 0, 1.0f);
    gemm(y1, u, W, nullptr, N1, F, F, F, F, 0, 0, 1.0f);
    // A12 = ef(x1d, A, L1d, at12); y1 += (A12 x1d) A
    gemm(Xf, x1d, Am, nullptr, N1, F, F, F, F, 0, 0, 1.0f);
    san_rowdot2<<<N1 / 8, 256, 0, stream>>>(Xf, at12, av, bv, N1);
    san_att_softmax<<<N1, 256, 0, stream>>>(Wb, L1d, av, bv, N1);
    gemm(u, Wb, x1d, nullptr, N1, F, N1, N1, F, 0, 0, 1.0f);
    gemm(y1, u, Am, y1, N1, F, F, F, F, 0, 0, 1.0f);
    // A13 = ef(z1, R, L1u, at13); y1 += (A13 z1) R
    gemm(Xf, z1, R, nullptr, N1, F, F, F, F, 0, 0, 1.0f);
    san_rowdot2<<<N1 / 8, 256, 0, stream>>>(Xf, at13, av, bv, N1);
    san_att_softmax<<<N1, 256, 0, stream>>>(Wb, L1u, av, bv, N1);
    gemm(u, Wb, z1, nullptr, N1, F, N1, N1, F, 0, 0, 1.0f);
    gemm(y1, u, R, y1, N1, F, F, F, F, 0, 0, 1.0f);
    // projection: t2 = (I - eps L1)^2 z1 with L1 = L1d + L1u; y1 += t2 Wh
    gemm(t1, L1d, z1, z1, N1, F, N1, N1, F, 0, 0, -EPS_P);
    gemm(t1, L1u, z1, t1, N1, F, N1, N1, F, 0, 0, -EPS_P);
    gemm(t2, L1d, t1, t1, N1, F, N1, N1, F, 0, 0, -EPS_P);
    gemm(t2, L1u, t1, t2, N1, F, N1, N1, F, 0, 0, -EPS_P);
    gemm(y1, t2, Wh, y1, N1, F, F, F, F, 0, 0, 1.0f);
    san_sigmoid<<<(N1 * F + 255) / 256, 256, 0, stream>>>(out + (size_t)N0 * F, y1, N1 * F);

    // ================= z0 pipeline =================
    gemm(y0, z0, W, nullptr, N0, F, F, F, F, 0, 0, 1.0f);
    gemm(u, B1, z1, nullptr, N0, F, N1, N1, F, 0, 0, 1.0f);      // B1 z1
    gemm(y0, u, Am, y0, N0, F, F, F, F, 0, 0, 1.0f);
    gemm(t1, L0, z0, z0, N0, F, N0, N0, F, 0, 0, -EPS_P);
    gemm(t2, L0, t1, t1, N0, F, N0, N0, F, 0, 0, -EPS_P);
    gemm(y0, t2, Wh, y0, N0, F, F, F, F, 0, 0, 1.0f);
    san_sigmoid<<<(N0 * F + 255) / 256, 256, 0, stream>>>(out, y0, N0 * F);

    // ================= z2 pipeline =================
    // A21 = ef(z2, R, L2, at21); y2 = (A21 z2) W
    gemm(Xf, z2, R, nullptr, N2, F, F, F, F, 0, 0, 1.0f);
    san_rowdot2<<<N2 / 8, 256, 0, stream>>>(Xf, at21, av, bv, N2);
    san_att_softmax<<<N2, 256, 0, stream>>>(Wb, L2, av, bv, N2);
    gemm(u, Wb, z2, nullptr, N2, F, N2, N2, F, 0, 0, 1.0f);
    gemm(y2, u, W, nullptr, N2, F, F, F, F, 0, 0, 1.0f);
    // A22 = ef(x2u, D, L2, at22); y2 += (A22 x2u) R
    gemm(Xf, x2u, Dm, nullptr, N2, F, F, F, F, 0, 0, 1.0f);
    san_rowdot2<<<N2 / 8, 256, 0, stream>>>(Xf, at22, av, bv, N2);
    san_att_softmax<<<N2, 256, 0, stream>>>(Wb, L2, av, bv, N2);
    gemm(u, Wb, x2u, nullptr, N2, F, N2, N2, F, 0, 0, 1.0f);
    gemm(y2, u, R, y2, N2, F, F, F, F, 0, 0, 1.0f);
    // projection + output transform
    gemm(t1, L2, z2, z2, N2, F, N2, N2, F, 0, 0, -EPS_P);
    gemm(t2, L2, t1, t1, N2, F, N2, N2, F, 0, 0, -EPS_P);
    gemm(y2, t2, Wh, y2, N2, F, F, F, F, 0, 0, 1.0f);
    san_sigmoid<<<(N2 * F + 255) / 256, 256, 0, stream>>>(
        out + (size_t)N0 * F + (size_t)N1 * F, y2, N2 * F);
}